// OutputBlock_36189394436939
// MI455X (gfx1250) — compile-verified
//
#include <hip/hip_runtime.h>
#include <hip/hip_bf16.h>

#define HIDDEN 256
#define NRAD 16

typedef float v2f __attribute__((ext_vector_type(2)));
typedef float v8f __attribute__((ext_vector_type(8)));
typedef float v4f __attribute__((ext_vector_type(4)));

// ---------------------------------------------------------------------------
// Zero the scatter accumulator (ws is poisoned by the harness each run).
// ---------------------------------------------------------------------------
__global__ void zero_f32(float* __restrict__ p, int n4) {
    v4f z = {};
    for (int i = blockIdx.x * blockDim.x + threadIdx.x; i < n4;
         i += gridDim.x * blockDim.x) {
        ((v4f*)p)[i] = z;
    }
}

// ---------------------------------------------------------------------------
// Phase 1+2: g = (rbf @ W_rbf) * x, scatter-add into h[node][256].
// One thread == one channel (blockDim == 256). W_rbf column lives in 16
// registers; rbf rows for 32 edges staged through LDS per batch.
// Memory-bound: ~850 MB streamed; atomics land in the 192MB L2 (h = 51MB).
// ---------------------------------------------------------------------------
__global__ void edge_gate_scatter(const float* __restrict__ x,
                                  const float* __restrict__ rbf,
                                  const int*   __restrict__ idx,
                                  const float* __restrict__ Wrbf,
                                  float* __restrict__ h,
                                  int E) {
    __shared__ float s_rbf[32 * NRAD];  // 2 KB
    __shared__ int   s_idx[32];

    const int tid = threadIdx.x;  // channel 0..255

    float wcol[NRAD];
#pragma unroll
    for (int r = 0; r < NRAD; ++r) wcol[r] = Wrbf[r * HIDDEN + tid];

    const int batches = (E + 31) >> 5;
    for (int b = blockIdx.x; b < batches; b += gridDim.x) {
        const int e0 = b << 5;
        const int nE = min(32, E - e0);

        __syncthreads();  // protect LDS reuse from previous batch
        {
            const float* src = rbf + (size_t)e0 * NRAD;
            const int total = nE * NRAD;
            for (int j = tid; j < total; j += 256) s_rbf[j] = src[j];
            if (tid < nE) s_idx[tid] = idx[e0 + tid];
        }
        __syncthreads();

        for (int ee = 0; ee < nE; ++ee) {
            const int e = e0 + ee;
            float dot = 0.f;
#pragma unroll
            for (int r = 0; r < NRAD; ++r)
                dot = fmaf(s_rbf[ee * NRAD + r], wcol[r], dot);
            const float g = dot * x[(size_t)e * HIDDEN + tid];
            atomicAdd(&h[(size_t)s_idx[ee] * HIDDEN + tid], g);
        }
    }
}

// ---------------------------------------------------------------------------
// Phase 3+4: Out[M,N] = act(A[M,256] @ W[256,N] + bias) with
// V_WMMA_F32_16X16X4_F32. Each wave owns an 80x32 macro-tile
// (MT=5 m-tiles x NT=2 n-tiles): per K-step 4 B-loads (b32) + 5 A-loads
// (b64, all immediate offsets off ONE base) feed 10 WMMAs.
// B loads issue FIRST so A-loads stay in flight under the WMMA stream
// (in-order load retirement -> staircase s_wait_loadcnt).
// Fragment layouts per CDNA5 ISA 7.12.2 (fp32):
//   A 16x4 : lanes 0-15 row M, VGPR0/1 = K0/K1; lanes 16-31 = K2/K3
//   B 4x16 : lanes 0-15 col N, VGPR0/1 = K0/K1; lanes 16-31 = K2/K3
//   C/D    : lane&15 = col, VGPR r = row r (lanes<16) / r+8 (lanes>=16)
// ---------------------------------------------------------------------------
#define MT 5
#define NT 2

template <int N, bool ACT>
__global__ void mlp_wmma(const float* __restrict__ A,
                         const float* __restrict__ W,
                         const float* __restrict__ bias,
                         float* __restrict__ Out,
                         int M) {
    const int lane = threadIdx.x & 31;
    const int wave = threadIdx.x >> 5;

    const int nblk = N / (16 * NT);            // N macro-blocks
    const int mblk = M / (16 * MT);            // M macro-blocks
    const int w = blockIdx.x * (blockDim.x >> 5) + wave;
    if (w >= mblk * nblk) return;              // wave-uniform: EXEC all-1s
    const int mb = w / nblk;
    const int nb = w - mb * nblk;

    const int half = lane >> 4;                // K-pair select (0 or 1)
    const int l15  = lane & 15;

    // ONE A base; the 5 m-tile rows are compile-time byte offsets i*16384.
    const float* abase = A + (size_t)(mb * MT * 16 + l15) * HIDDEN + 2 * half;
    // ONE B base; n-tile cols are compile-time byte offsets j*64.
    const float* bbase = W + (size_t)(2 * half) * N + (nb * NT * 16 + l15);

    v8f c[MT][NT];
#pragma unroll
    for (int i = 0; i < MT; ++i)
#pragma unroll
        for (int j = 0; j < NT; ++j) c[i][j] = (v8f){};

#pragma unroll 2
    for (int k = 0; k < HIDDEN; k += 4) {
        // B first (needed by the first WMMA), then A: later A-loads remain
        // outstanding while early WMMAs run.
        v2f b[NT];
#pragma unroll
        for (int j = 0; j < NT; ++j) {
            b[j].x = bbase[(size_t)k * N + j * 16];      // W[k+2h  ][col]
            b[j].y = bbase[(size_t)k * N + N + j * 16];  // W[k+2h+1][col]
        }
        v2f a[MT];
#pragma unroll
        for (int i = 0; i < MT; ++i)
            a[i] = *(const v2f*)(abase + k + i * (16 * HIDDEN));

#pragma unroll
        for (int i = 0; i < MT; ++i)
#pragma unroll
            for (int j = 0; j < NT; ++j)
                c[i][j] = __builtin_amdgcn_wmma_f32_16x16x4_f32(
                    false, a[i], false, b[j], (short)0, c[i][j], false, false);
    }

    // Epilogue: bias + swish (fast rcp, no IEEE divide), strided stores.
#pragma unroll
    for (int j = 0; j < NT; ++j) {
        const int col = (nb * NT + j) * 16 + l15;
        float bb = 0.f;
        if (bias) bb = bias[col];
#pragma unroll
        for (int i = 0; i < MT; ++i) {
            float* orow = Out + (size_t)((mb * MT + i) * 16 + 8 * half) * N + col;
#pragma unroll
            for (int r = 0; r < 8; ++r) {
                float v = c[i][j][r] + bb;
                if (ACT)
                    v = v * __builtin_amdgcn_rcpf(1.f + __expf(-v));  // swish
                orow[(size_t)r * N] = v;
            }
        }
    }
}

// ---------------------------------------------------------------------------
// d_in order: x[E,256], rbf[E,16], i[E], num_nodes, W_rbf[16,256],
//             Ws[3,256,256], bs[3,256], W_out[256,128]
// ---------------------------------------------------------------------------
extern "C" void kernel_launch(void* const* d_in, const int* in_sizes, int n_in,
                              void* d_out, int out_size, void* d_ws, size_t ws_size,
                              hipStream_t stream) {
    const float* x     = (const float*)d_in[0];
    const float* rbf   = (const float*)d_in[1];
    const int*   idx   = (const int*)d_in[2];
    const float* Wrbf  = (const float*)d_in[4];
    const float* Ws    = (const float*)d_in[5];
    const float* bs    = (const float*)d_in[6];
    const float* Wout  = (const float*)d_in[7];
    float*       out   = (float*)d_out;

    const int E  = in_sizes[2];        // edge count
    const int NN = out_size / 128;     // num_nodes (out is [NN,128] f32)

    float* h0 = (float*)d_ws;                      // [NN,256]
    float* h1 = h0 + (size_t)NN * HIDDEN;          // [NN,256]

    // 1) zero accumulator
    {
        const int n4 = (NN * HIDDEN) / 4;
        zero_f32<<<4096, 256, 0, stream>>>(h0, n4);
    }

    // 2) edge gate + scatter
    {
        const int batches = (E + 31) / 32;
        const int grid = batches < 16384 ? batches : 16384;
        edge_gate_scatter<<<grid, 256, 0, stream>>>(x, rbf, idx, Wrbf, h0, E);
    }

    // 3) hidden layers (swish), ping-pong h0 <-> h1; 8 waves / block
    {
        const int waves = (NN / (16 * MT)) * (HIDDEN / (16 * NT));
        const int grid = (waves + 7) / 8;
        mlp_wmma<HIDDEN, true><<<grid, 256, 0, stream>>>(
            h0, Ws + 0 * HIDDEN * HIDDEN, bs + 0 * HIDDEN, h1, NN);
        mlp_wmma<HIDDEN, true><<<grid, 256, 0, stream>>>(
            h1, Ws + 1 * HIDDEN * HIDDEN, bs + 1 * HIDDEN, h0, NN);
        mlp_wmma<HIDDEN, true><<<grid, 256, 0, stream>>>(
            h0, Ws + 2 * HIDDEN * HIDDEN, bs + 2 * HIDDEN, h1, NN);
    }

    // 4) output projection (no bias, no activation) -> d_out
    {
        const int waves = (NN / (16 * MT)) * (128 / (16 * NT));
        const int grid = (waves + 7) / 8;
        mlp_wmma<128, false><<<grid, 256, 0, stream>>>(
            h1, Wout, nullptr, out, NN);
    }
}